// MeanAggregator_3075196584045
// MI455X (gfx1250) — compile-verified
//
#include <hip/hip_runtime.h>

// MeanAggregator for MI455X (gfx1250):
//   out[b, :] = (1/25) * sum_s features[neigh_idx[b*25+s], :]
// B=65536, S=25, D=128 (fp32). Pure gather-bandwidth problem (~0.25 flop/B):
// ~840 MB gathered + 33.5 MB stored -> ~37 us floor at 23.3 TB/s, less with
// L2 hits (256 MB feature table vs 192 MB L2, ~3.3x mean row reuse).
//
// Mapping: one wave32 per batch node. Each feature row is 128 f32 = 512 B =
// 32 lanes x 16 B, so ONE global_load_async_to_lds_b128 moves one whole row.
// 25 async gathers are issued back-to-back (ASYNCcnt tracks them), then the
// wave drains with s_wait_asynccnt 0 and reduces the 25 rows out of LDS with
// ds_load_b128, scales by 1/25, and writes one coalesced non-temporal b128
// store (write-once output must not evict the hot feature table from L2).

#define S_FANOUT 25
#define D_FEAT   128
#define WAVES_PB 4                      // waves per block
#define BLOCK_T  (WAVES_PB * 32)        // 128 threads
#define SLAB_F   (S_FANOUT * D_FEAT)    // floats per wave slab (12800 B)

typedef __attribute__((ext_vector_type(4))) float v4f;

__device__ __forceinline__ void async_row_b128(const float* gsrc, float* ldst) {
  // LDS byte address = low 32 bits of the generic pointer (ISA: addr[31:0]).
  const unsigned lofs = (unsigned)(unsigned long long)ldst;
  // GV mode: vdst = LDS addr VGPR, vaddr = 64-bit global addr VGPR pair.
  asm volatile("global_load_async_to_lds_b128 %0, %1, off"
               :: "v"(lofs), "v"(gsrc)
               : "memory");
}

__device__ __forceinline__ void wait_async0() {
  asm volatile("s_wait_asynccnt 0" ::: "memory");
}

__global__ __launch_bounds__(BLOCK_T) void mean_agg_gather_async(
    const float* __restrict__ feat,        // [N_NODES, 128]
    const long long* __restrict__ nidx,    // [B, 25] int64
    float* __restrict__ out,               // [B, 128]
    int batch) {
  __shared__ __align__(16) float lds[WAVES_PB * SLAB_F];  // 51200 B

  const int lane = threadIdx.x & 31;
  const int wave = threadIdx.x >> 5;
  float* slab = &lds[wave * SLAB_F];

  const int wave_gid = blockIdx.x * WAVES_PB + wave;
  const int nwaves   = gridDim.x * WAVES_PB;

  for (int b = wave_gid; b < batch; b += nwaves) {
    // b is wave-uniform; make it provably so -> neighbor ids become s_load.
    const int bu = __builtin_amdgcn_readfirstlane(b);
    const long long* __restrict__ ip = nidx + (long long)bu * S_FANOUT;

    // Issue 25 async row gathers: row ip[s] -> slab[s*128 ..]. Each lane
    // supplies bytes [16*lane, 16*lane+16) of the row.
#pragma unroll
    for (int s = 0; s < S_FANOUT; ++s) {
      const long long row = ip[s];                               // scalar load
      const float* g = feat + row * (long long)D_FEAT + lane * 4;
      async_row_b128(g, slab + s * D_FEAT + lane * 4);
    }

    wait_async0();

    // Reduce 25 rows out of LDS (ds_load_b128 per row), scale.
    v4f acc = (v4f)(0.f);
#pragma unroll
    for (int s = 0; s < S_FANOUT; ++s) {
      const v4f v = *(const v4f*)(slab + s * D_FEAT + lane * 4);
      acc += v;
    }
    acc *= (1.0f / (float)S_FANOUT);

    // Non-temporal b128 store: output is streamed once, keep it out of L2.
    __builtin_nontemporal_store(acc, (v4f*)(out + (long long)b * D_FEAT + lane * 4));
  }
}

extern "C" void kernel_launch(void* const* d_in, const int* in_sizes, int n_in,
                              void* d_out, int out_size, void* d_ws, size_t ws_size,
                              hipStream_t stream) {
  (void)n_in; (void)d_ws; (void)ws_size; (void)out_size;
  const float*     feat = (const float*)d_in[0];
  const long long* nidx = (const long long*)d_in[1];
  float*           out  = (float*)d_out;

  const int batch = in_sizes[1] / S_FANOUT;          // 65536
  const int blocks = (batch + WAVES_PB - 1) / WAVES_PB;

  mean_agg_gather_async<<<blocks, BLOCK_T, 0, stream>>>(feat, nidx, out, batch);
}